// BatchAugment_52003464020158
// MI455X (gfx1250) — compile-verified
//
#include <hip/hip_runtime.h>
#include <hip/hip_bf16.h>
#include <math.h>

typedef __attribute__((ext_vector_type(2))) float v2f;
typedef __attribute__((ext_vector_type(8))) float v8f;

#define B_  64
#define C_  3
#define H_  384
#define W_  384
#define HW_ (H_ * W_)

// ---------------------------------------------------------------------------
// Pass 1: flip + bilinear rotate + brightness + clip.  One thread per (b,y,x),
// handles all 3 channels (shared gather geometry). Writes intermediate to out.
// ---------------------------------------------------------------------------
__global__ void __launch_bounds__(256)
aug_pass1(const float* __restrict__ x,
          const unsigned char* __restrict__ hflip,
          const unsigned char* __restrict__ vflip,
          const unsigned char* __restrict__ rotm,
          const float* __restrict__ angles,
          const float* __restrict__ bright,
          float* __restrict__ out)
{
    int t = blockIdx.x * blockDim.x + threadIdx.x;
    if (t >= B_ * HW_) return;
    int b  = t / HW_;
    int r  = t - b * HW_;
    int yo = r / W_;
    int xo = r - yo * W_;

    const bool hf = hflip[b] != 0;
    const bool vf = vflip[b] != 0;
    const bool rm = rotm[b]  != 0;
    const float br = bright[b];

    const float* __restrict__ xb = x + (size_t)b * C_ * HW_;
    float vals[C_];

    if (rm) {
        float th = angles[b] * 0.017453292519943295f;   // deg2rad
        float cth = cosf(th), sth = sinf(th);
        float GX = (2.0f * (float)xo + 1.0f) * (1.0f / (float)W_) - 1.0f;
        float GY = (2.0f * (float)yo + 1.0f) * (1.0f / (float)H_) - 1.0f;
        float xin = cth * GX - sth * GY;
        float yin = sth * GX + cth * GY;
        float ix = ((xin + 1.0f) * (float)W_ - 1.0f) * 0.5f;
        float iy = ((yin + 1.0f) * (float)H_ - 1.0f) * 0.5f;
        float ix0 = floorf(ix), iy0 = floorf(iy);
        float ix1 = ix0 + 1.0f, iy1 = iy0 + 1.0f;
        float wx1 = ix - ix0, wx0 = 1.0f - wx1;
        float wy1 = iy - iy0, wy0 = 1.0f - wy1;

        // validity on un-clipped coords (matches reference), clip, then flip.
        bool vX0 = (ix0 >= 0.0f) & (ix0 < (float)W_);
        bool vX1 = (ix1 >= 0.0f) & (ix1 < (float)W_);
        bool vY0 = (iy0 >= 0.0f) & (iy0 < (float)H_);
        bool vY1 = (iy1 >= 0.0f) & (iy1 < (float)H_);
        int cx0 = (int)fminf(fmaxf(ix0, 0.0f), (float)(W_ - 1));
        int cx1 = (int)fminf(fmaxf(ix1, 0.0f), (float)(W_ - 1));
        int cy0 = (int)fminf(fmaxf(iy0, 0.0f), (float)(H_ - 1));
        int cy1 = (int)fminf(fmaxf(iy1, 0.0f), (float)(H_ - 1));
        if (hf) { cx0 = W_ - 1 - cx0; cx1 = W_ - 1 - cx1; }
        if (vf) { cy0 = H_ - 1 - cy0; cy1 = H_ - 1 - cy1; }

        float w00 = wy0 * wx0 * (float)(vY0 & vX0);
        float w01 = wy0 * wx1 * (float)(vY0 & vX1);
        float w10 = wy1 * wx0 * (float)(vY1 & vX0);
        float w11 = wy1 * wx1 * (float)(vY1 & vX1);

        int i00 = cy0 * W_ + cx0;
        int i01 = cy0 * W_ + cx1;
        int i10 = cy1 * W_ + cx0;
        int i11 = cy1 * W_ + cx1;
#pragma unroll
        for (int c = 0; c < C_; ++c) {
            const float* pl = xb + c * HW_;
            vals[c] = pl[i00] * w00 + pl[i01] * w01 + pl[i10] * w10 + pl[i11] * w11;
        }
    } else {
        int xi = hf ? (W_ - 1 - xo) : xo;
        int yi = vf ? (H_ - 1 - yo) : yo;
        int idx = yi * W_ + xi;
#pragma unroll
        for (int c = 0; c < C_; ++c) vals[c] = xb[c * HW_ + idx];
    }

    float* ob = out + (size_t)b * C_ * HW_ + r;
#pragma unroll
    for (int c = 0; c < C_; ++c) {
        float v = vals[c] * br;
        ob[c * HW_] = fminf(fmaxf(v, 0.0f), 1.0f);
    }
}

// ---------------------------------------------------------------------------
// Pass 1b: per-(b,c) plane mean via V_WMMA_F32_16X16X4_F32 with all-ones B.
// D = A*ones + C  => each WMMA reduces 64 f32 elements (wave-wide) exactly.
// One 256-thread block (8 waves) per plane; EXEC all-ones (WMMA requirement).
// Scalar (compile-time) trip count + 4 independent accumulator chains so the
// 4 global_load_b64 per unrolled body issue as one clause and the 4 WMMAs run
// back-to-back (D->C chaining carries no WMMA->WMMA hazard per ISA 7.12.1).
// ---------------------------------------------------------------------------
__global__ void __launch_bounds__(256)
plane_mean_wmma(const float* __restrict__ inter, float* __restrict__ means)
{
    const int plane = blockIdx.x;              // b*3 + c, 0..191
    const int lane = threadIdx.x & 31;
    const int wave = threadIdx.x >> 5;
    // wave's first element; consecutive wave-iterations stride 8 waves * 64.
    const float* __restrict__ p =
        inter + (size_t)plane * HW_ + wave * 64 + lane * 2;

    float waveSum;
#if __has_builtin(__builtin_amdgcn_wmma_f32_16x16x4_f32)
    v8f acc0 = {}, acc1 = {}, acc2 = {}, acc3 = {};
    v2f ones; ones.x = 1.0f; ones.y = 1.0f;
    // HW_/64 = 2304 chunks; /8 waves = 288 per wave; unroll x4 -> 72 iters.
    constexpr int ITER   = HW_ / 64 / 8;   // 288 (uniform, compile-time)
    constexpr int STRIDE = 8 * 64;         // floats between wave-iterations
#pragma unroll 1
    for (int i = 0; i < ITER; i += 4) {
        v2f a0 = *(const v2f*)(p + (size_t)(i + 0) * STRIDE);
        v2f a1 = *(const v2f*)(p + (size_t)(i + 1) * STRIDE);
        v2f a2 = *(const v2f*)(p + (size_t)(i + 2) * STRIDE);
        v2f a3 = *(const v2f*)(p + (size_t)(i + 3) * STRIDE);
        acc0 = __builtin_amdgcn_wmma_f32_16x16x4_f32(false, a0, false, ones,
                                                     (short)0, acc0, false, false);
        acc1 = __builtin_amdgcn_wmma_f32_16x16x4_f32(false, a1, false, ones,
                                                     (short)0, acc1, false, false);
        acc2 = __builtin_amdgcn_wmma_f32_16x16x4_f32(false, a2, false, ones,
                                                     (short)0, acc2, false, false);
        acc3 = __builtin_amdgcn_wmma_f32_16x16x4_f32(false, a3, false, ones,
                                                     (short)0, acc3, false, false);
    }
    float s = 0.0f;
#pragma unroll
    for (int k = 0; k < 8; ++k)
        s += (acc0[k] + acc1[k]) + (acc2[k] + acc3[k]);
    // All 16 columns hold identical row-sums -> full-wave sum = 16 * total.
    for (int off = 16; off > 0; off >>= 1) s += __shfl_xor(s, off, 32);
    waveSum = s * (1.0f / 16.0f);
#else
    float s = 0.0f;
    for (int i = 0; i < HW_ / 64 / 8; ++i) {
        const float* q = p + (size_t)i * 8 * 64;
        s += q[0] + q[1];
    }
    for (int off = 16; off > 0; off >>= 1) s += __shfl_xor(s, off, 32);
    waveSum = s;
#endif

    __shared__ float wsum[8];
    if (lane == 0) wsum[wave] = waveSum;
    __syncthreads();
    if (threadIdx.x == 0) {
        float tot = 0.0f;
        for (int w = 0; w < 8; ++w) tot += wsum[w];
        means[plane] = tot * (1.0f / (float)HW_);
    }
}

// ---------------------------------------------------------------------------
// Pass 2: contrast (around per-(b,c) mean) + clip + hue rotation, in place.
// ---------------------------------------------------------------------------
__global__ void __launch_bounds__(256)
aug_pass2(float* __restrict__ io,
          const float* __restrict__ means,
          const float* __restrict__ contrast,
          const float* __restrict__ hue)
{
    int t = blockIdx.x * blockDim.x + threadIdx.x;
    if (t >= B_ * HW_) return;
    int b = t / HW_;
    int r = t - b * HW_;

    const float ct = contrast[b];
    const float hu = hue[b];
    float m0 = means[b * 3 + 0];
    float m1 = means[b * 3 + 1];
    float m2 = means[b * 3 + 2];

    size_t base = (size_t)b * C_ * HW_ + r;
    float rr = io[base];
    float gg = io[base + HW_];
    float bb = io[base + 2 * HW_];

    rr = fminf(fmaxf((rr - m0) * ct + m0, 0.0f), 1.0f);
    gg = fminf(fmaxf((gg - m1) * ct + m1, 0.0f), 1.0f);
    bb = fminf(fmaxf((bb - m2) * ct + m2, 0.0f), 1.0f);

    // RGB -> H,S,V
    float maxc = fmaxf(rr, fmaxf(gg, bb));
    float minc = fminf(rr, fminf(gg, bb));
    float v  = maxc;
    float dc = maxc - minc;
    float sat = (maxc == 0.0f) ? 0.0f : dc / fmaxf(maxc, 1e-12f);
    float dcs = (dc == 0.0f) ? 1.0f : dc;
    float rc = (maxc - rr) / dcs;
    float gc = (maxc - gg) / dcs;
    float bc = (maxc - bb) / dcs;
    float h = (maxc == rr) ? (bc - gc)
            : ((maxc == gg) ? (2.0f + rc - bc) : (4.0f + gc - rc));
    if (dc == 0.0f) h = 0.0f;
    h = h * (1.0f / 6.0f);
    h -= floorf(h);          // mod 1
    h += hu;
    h -= floorf(h);          // mod 1

    float h6 = h * 6.0f;
    float fi = floorf(h6);
    float f  = h6 - fi;
    int   i  = ((int)fi) % 6;
    if (i < 0) i += 6;
    float p = v * (1.0f - sat);
    float q = v * (1.0f - f * sat);
    float tt = v * (1.0f - (1.0f - f) * sat);

    float ro, go, bo;
    switch (i) {
        case 0: ro = v;  go = tt; bo = p;  break;
        case 1: ro = q;  go = v;  bo = p;  break;
        case 2: ro = p;  go = v;  bo = tt; break;
        case 3: ro = p;  go = q;  bo = v;  break;
        case 4: ro = tt; go = p;  bo = v;  break;
        default: ro = v; go = p;  bo = q;  break;
    }

    io[base]           = ro;
    io[base + HW_]     = go;
    io[base + 2 * HW_] = bo;
}

// ---------------------------------------------------------------------------
extern "C" void kernel_launch(void* const* d_in, const int* in_sizes, int n_in,
                              void* d_out, int out_size, void* d_ws, size_t ws_size,
                              hipStream_t stream)
{
    const float*         x   = (const float*)d_in[0];
    const unsigned char* hf  = (const unsigned char*)d_in[1]; // jax bool = 1B
    const unsigned char* vf  = (const unsigned char*)d_in[2];
    const unsigned char* rm  = (const unsigned char*)d_in[3];
    const float*         ang = (const float*)d_in[4];
    const float*         br  = (const float*)d_in[5];
    const float*         ctr = (const float*)d_in[6];
    const float*         hu  = (const float*)d_in[7];
    float* out   = (float*)d_out;
    float* means = (float*)d_ws;   // 192 floats

    const int nPix = B_ * HW_;
    const int threads = 256;
    const int blocks = (nPix + threads - 1) / threads;

    aug_pass1<<<blocks, threads, 0, stream>>>(x, hf, vf, rm, ang, br, out);
    plane_mean_wmma<<<B_ * C_, threads, 0, stream>>>(out, means);
    aug_pass2<<<blocks, threads, 0, stream>>>(out, means, ctr, hu);
}